// GraphConvolution_48567490183787
// MI455X (gfx1250) — compile-verified
//
#include <hip/hip_runtime.h>
#include <hip/hip_bf16.h>

typedef __attribute__((ext_vector_type(2))) float v2f;
typedef __attribute__((ext_vector_type(8))) float v8f;

#define D 128  // D_IN == D_OUT == 128

// ---------------------------------------------------------------------------
// Kernel 1: support = x @ W using V_WMMA_F32_16X16X4_F32 (exact fp32 WMMA).
// Block = 256 threads = 8 waves. W (64KB) is staged into LDS once per block
// (WGP has 320KB LDS), then each wave computes one 16x16 tile of a 16-row
// strip: A from pipelined global_load_b64, B from ds_load_b32, K=128 stepped
// by 4 -> 32 v_wmma_f32_16x16x4_f32 per wave.
// ---------------------------------------------------------------------------
__global__ __launch_bounds__(256) void gcn_gemm_wmma(
    const float* __restrict__ x, const float* __restrict__ w,
    float* __restrict__ support, int n_nodes) {
  __shared__ float wlds[D * D];        // 64 KB: full 128x128 W tile, row-major

  // Stage W: 256 threads x 16 float4 each = 128*128 floats.
  for (int idx = threadIdx.x; idx < (D * D) / 4; idx += 256) {
    const int r  = idx >> 5;           // row 0..127
    const int c4 = (idx & 31) << 2;    // col 0,4,...,124
    *(float4*)(&wlds[r * D + c4]) = *(const float4*)(w + r * D + c4);
  }
  __syncthreads();                     // s_barrier_signal / s_barrier_wait

  const int wave = threadIdx.x >> 5;   // 0..7 -> N tile
  const int lane = threadIdx.x & 31;
  const int m0   = blockIdx.x * 16;
  const int n0   = wave * 16;
  const int mi   = lane & 15;          // M (for A) / N (for B) sub-index
  const int kh   = (lane >> 4) << 1;   // K half-select: 0 or 2

  int row = m0 + mi;
  if (row >= n_nodes) row = n_nodes - 1;   // clamp (50000 % 16 == 0 anyway)
  const float* __restrict__ xrow = x + (size_t)row * D;

  v8f acc = {};
#pragma unroll
  for (int k = 0; k < D; k += 4) {
    // A 16x4 tile of x: VGPR0 -> K = k+kh, VGPR1 -> K = k+kh+1 (8B-aligned pair)
    v2f a = *(const v2f*)(xrow + k + kh);
    // B 4x16 tile of W from LDS: lanes carry N, same K half split as A
    v2f b;
    b.x = wlds[(k + kh) * D + n0 + mi];
    b.y = wlds[(k + kh + 1) * D + n0 + mi];
    // (neg_a, A, neg_b, B, c_mod, C, reuse_a, reuse_b)
    acc = __builtin_amdgcn_wmma_f32_16x16x4_f32(
        false, a, false, b, (short)0, acc, false, false);
  }

  // C/D layout: VGPR j, lanes 0-15 -> M=j, lanes 16-31 -> M=j+8; N = lane&15
  const int mbase = m0 + ((lane >> 4) << 3);
#pragma unroll
  for (int j = 0; j < 8; ++j) {
    const int r = mbase + j;
    if (r < n_nodes) support[(size_t)r * D + n0 + mi] = acc[j];
  }
}

// ---------------------------------------------------------------------------
// Kernel 2: out[i, :] = bias[:]  (also clears the 0xAA poison deterministically)
// ---------------------------------------------------------------------------
__global__ void gcn_bias_init(float* __restrict__ out,
                              const float* __restrict__ bias, int total) {
  const int i = blockIdx.x * blockDim.x + threadIdx.x;
  if (i < total) out[i] = bias[i & (D - 1)];
}

// ---------------------------------------------------------------------------
// Kernel 3: edge scatter-add (the bottleneck; L2-atomic-throughput bound).
// One wave32 per edge: lane l owns features [4l, 4l+4) -> coalesced 512B
// gather from support[src], then 4 native global_atomic_add_f32 into
// out[dst]. support & out are L2-resident (25.6MB each vs 192MB L2).
// ---------------------------------------------------------------------------
__global__ __launch_bounds__(256) void gcn_edge_scatter(
    const float* __restrict__ support, const int* __restrict__ esrc,
    const int* __restrict__ edst, const float* __restrict__ eval,
    float* __restrict__ out, int n_edges) {
  const int lane   = threadIdx.x & 31;
  const int gwave  = (int)((blockIdx.x * blockDim.x + threadIdx.x) >> 5);
  const int nwaves = (int)((gridDim.x * blockDim.x) >> 5);

  for (int e = gwave; e < n_edges; e += nwaves) {
    const int   s = esrc[e];
    const int   d = edst[e];
    const float v = eval[e];
    const float4 g = ((const float4*)(support + (size_t)s * D))[lane];
    float* op = out + (size_t)d * D + lane * 4;
    unsafeAtomicAdd(op + 0, g.x * v);   // -> global_atomic_add_f32
    unsafeAtomicAdd(op + 1, g.y * v);
    unsafeAtomicAdd(op + 2, g.z * v);
    unsafeAtomicAdd(op + 3, g.w * v);
  }
}

// ---------------------------------------------------------------------------
// Launch: GEMM (ws <- support), bias init, edge scatter. All on `stream`.
// Inputs: x, edge_src, edge_dst, edge_val, weight, bias.
// ---------------------------------------------------------------------------
extern "C" void kernel_launch(void* const* d_in, const int* in_sizes, int n_in,
                              void* d_out, int out_size, void* d_ws, size_t ws_size,
                              hipStream_t stream) {
  const float* x    = (const float*)d_in[0];
  const int*   esrc = (const int*)d_in[1];
  const int*   edst = (const int*)d_in[2];
  const float* eval = (const float*)d_in[3];
  const float* wgt  = (const float*)d_in[4];
  const float* bias = (const float*)d_in[5];
  float*       out  = (float*)d_out;
  float*       supp = (float*)d_ws;   // [n_nodes, 128] scratch

  const int n_nodes = in_sizes[0] / D;
  const int n_edges = in_sizes[1];

  const int gemm_blocks = (n_nodes + 15) / 16;   // 3125 for N=50000
  gcn_gemm_wmma<<<gemm_blocks, 256, 0, stream>>>(x, wgt, supp, n_nodes);

  const int init_blocks = (out_size + 255) / 256;
  gcn_bias_init<<<init_blocks, 256, 0, stream>>>(out, bias, out_size);

  // 8 waves/block; grid-stride so each wave handles ~8-10 edges
  const int scat_blocks = 12288;
  gcn_edge_scatter<<<scat_blocks, 256, 0, stream>>>(supp, esrc, edst, eval,
                                                    out, n_edges);
}